// HolographicAttentionNetwork_86887188398971
// MI455X (gfx1250) — compile-verified
//
#include <hip/hip_runtime.h>
#include <hip/hip_bf16.h>

// HolographicAttentionNetwork for MI455X (gfx1250, wave32, WMMA + async-to-LDS).
//
// Algebra:  q_b = key_b @ (key_b @ C_b)^T,  C_b = (sum_k W[:,k,:]) @ V_b  (128x128)
// Pipeline: k_w2 -> k_C -> k_mem -> k_q   (all GEMMs on V_WMMA_F32_16X16X4_F32)

typedef __attribute__((ext_vector_type(2))) float v2f;
typedef __attribute__((ext_vector_type(8))) float v8f;

#define B_ 32
#define L_ 2048
#define N_ 128   // KEY_DIM
#define D_ 64    // VALUE_DIM (summed away)
#define A_ 256   // N_ACTIONS

#define LDS_STRIDE 132   // 128 + 4 pad: conflict-free row-varying reads, 16B-aligned rows

static __device__ __forceinline__ v8f wmma4(v2f a, v2f b, v8f c) {
  // D(16x16,f32) = A(16x4,f32) * B(4x16,f32) + C
  return __builtin_amdgcn_wmma_f32_16x16x4_f32(
      /*neg_a=*/false, a, /*neg_b=*/false, b,
      /*c_mod=*/(short)0, c, /*reuse_a=*/false, /*reuse_b=*/false);
}

// Async 16B global -> LDS copy (per-lane addresses), tracked by ASYNCcnt.
static __device__ __forceinline__ void async_copy16(const float* gsrc, float* lds_dst) {
  const unsigned loff = (unsigned)(uintptr_t)lds_dst;  // low 32b of shared-aperture addr
  asm volatile("global_load_async_to_lds_b128 %0, %1, off"
               :
               : "v"(loff), "v"(gsrc)
               : "memory");
}

static __device__ __forceinline__ void async_wait0() {
  asm volatile("s_wait_asynccnt 0x0" ::: "memory");
}

// ---------------------------------------------------------------------------
// Kernel 1: W2[n][a] = sum_k W[n][k][a]    (W: 128 x 64 x 256)
// ---------------------------------------------------------------------------
__global__ __launch_bounds__(256) void k_w2(const float* __restrict__ W,
                                            float* __restrict__ W2) {
  const int idx = blockIdx.x * 256 + threadIdx.x;   // 0 .. 128*256-1
  const int n = idx >> 8;
  const int a = idx & 255;
  const float* p = W + (size_t)n * D_ * A_ + a;
  float s = 0.f;
#pragma unroll
  for (int k = 0; k < D_; ++k) s += p[(size_t)k * A_];
  W2[idx] = s;
}

// ---------------------------------------------------------------------------
// Kernel 2: C_b (128x128) = W2 (128x256) @ V_b (256x128).  One WG per batch.
// 8 waves, each wave does a 32x64 subtile; K=256 in steps of 4.
// ---------------------------------------------------------------------------
__global__ __launch_bounds__(256) void k_C(const float* __restrict__ W2,
                                           const float* __restrict__ V,
                                           float* __restrict__ C) {
  const int b  = blockIdx.x;
  const int t  = threadIdx.x & 31;
  const int w  = threadIdx.x >> 5;
  const int wr = w >> 1;          // row group: [wr*32, wr*32+32)
  const int wc = w & 1;           // col group: [wc*64, wc*64+64)
  const int ln = t & 15;
  const int kh = t >> 4;          // lanes 16-31 handle K+2,K+3

  const float* Vb = V + (size_t)b * A_ * N_;
  v8f acc[2][4] = {};

  for (int k0 = 0; k0 < A_; k0 += 4) {
    const int kr = k0 + 2 * kh;
    v2f af[2], bf[4];
#pragma unroll
    for (int i = 0; i < 2; ++i) {
      const float* p = W2 + (size_t)(wr * 32 + i * 16 + ln) * A_ + kr;
      af[i].x = p[0]; af[i].y = p[1];
    }
#pragma unroll
    for (int j = 0; j < 4; ++j) {
      const float* p = Vb + (size_t)kr * N_ + (wc * 64 + j * 16 + ln);
      bf[j].x = p[0]; bf[j].y = p[N_];
    }
#pragma unroll
    for (int i = 0; i < 2; ++i)
#pragma unroll
      for (int j = 0; j < 4; ++j)
        acc[i][j] = wmma4(af[i], bf[j], acc[i][j]);
  }

  float* Cb = C + (size_t)b * N_ * N_;
#pragma unroll
  for (int i = 0; i < 2; ++i)
#pragma unroll
    for (int j = 0; j < 4; ++j)
#pragma unroll
      for (int v = 0; v < 8; ++v) {
        const int m = wr * 32 + i * 16 + v + 8 * kh;
        const int n = wc * 64 + j * 16 + ln;
        Cb[(size_t)m * N_ + n] = acc[i][j][v];
      }
}

// ---------------------------------------------------------------------------
// Kernel 3: mem_b (2048x128) = key_b @ C_b.  WG = 128 rows x 128 cols, K=128.
// key tile + C staged in LDS via async-to-LDS copies.
// ---------------------------------------------------------------------------
__global__ __launch_bounds__(256) void k_mem(const float* __restrict__ K,
                                             const float* __restrict__ C,
                                             float* __restrict__ M) {
  extern __shared__ float sm[];
  float* Kt = sm;                      // 128 x LDS_STRIDE (padded)
  float* Ct = sm + 128 * LDS_STRIDE;   // 128 x 128 (unpadded: row-wise reads)

  const int b   = blockIdx.y;
  const int l0  = blockIdx.x * 128;
  const int tid = threadIdx.x;
  const float* Kb = K + ((size_t)b * L_ + l0) * N_;
  const float* Cb = C + (size_t)b * N_ * N_;

#pragma unroll
  for (int it = 0; it < 16; ++it) {     // 4096 x 16B = 128x128 floats each
    const int c = tid + it * 256;
    const int r = c >> 5, q4 = c & 31;
    async_copy16(Kb + (size_t)r * N_ + q4 * 4, Kt + r * LDS_STRIDE + q4 * 4);
    async_copy16(Cb + (size_t)c * 4, Ct + (size_t)c * 4);
  }
  async_wait0();
  __syncthreads();

  const int t  = tid & 31, w = tid >> 5;
  const int wr = w >> 1, wc = w & 1;
  const int ln = t & 15, kh = t >> 4;

  v8f acc[2][4] = {};
  for (int k0 = 0; k0 < N_; k0 += 4) {
    const int kr = k0 + 2 * kh;
    v2f af[2], bf[4];
#pragma unroll
    for (int i = 0; i < 2; ++i)
      af[i] = *(const v2f*)(Kt + (wr * 32 + i * 16 + ln) * LDS_STRIDE + kr);
#pragma unroll
    for (int j = 0; j < 4; ++j) {
      const int n = wc * 64 + j * 16 + ln;
      bf[j].x = Ct[kr * N_ + n];
      bf[j].y = Ct[(kr + 1) * N_ + n];
    }
#pragma unroll
    for (int i = 0; i < 2; ++i)
#pragma unroll
      for (int j = 0; j < 4; ++j)
        acc[i][j] = wmma4(af[i], bf[j], acc[i][j]);
  }

  float* Mb = M + ((size_t)b * L_ + l0) * N_;
#pragma unroll
  for (int i = 0; i < 2; ++i)
#pragma unroll
    for (int j = 0; j < 4; ++j)
#pragma unroll
      for (int v = 0; v < 8; ++v) {
        const int lr = wr * 32 + i * 16 + v + 8 * kh;
        const int dc = wc * 64 + j * 16 + ln;
        Mb[(size_t)lr * N_ + dc] = acc[i][j][v];
      }
}

// ---------------------------------------------------------------------------
// Kernel 4: q_b (2048x2048) = key_b @ mem_b^T.  WG = 128x128 output tile,
// full K=128 staged in LDS once via async-to-LDS. 256 WMMAs per wave.
// ---------------------------------------------------------------------------
__global__ __launch_bounds__(256) void k_q(const float* __restrict__ K,
                                           const float* __restrict__ M,
                                           float* __restrict__ Q) {
  extern __shared__ float sm[];
  float* Kt = sm;                      // 128 x LDS_STRIDE
  float* Mt = sm + 128 * LDS_STRIDE;   // 128 x LDS_STRIDE

  const int b   = blockIdx.z;
  const int l0  = blockIdx.y * 128;
  const int m0  = blockIdx.x * 128;
  const int tid = threadIdx.x;
  const float* Kb = K + ((size_t)b * L_ + l0) * N_;
  const float* Mb = M + ((size_t)b * L_ + m0) * N_;

#pragma unroll
  for (int it = 0; it < 16; ++it) {
    const int c = tid + it * 256;
    const int r = c >> 5, q4 = c & 31;
    async_copy16(Kb + (size_t)r * N_ + q4 * 4, Kt + r * LDS_STRIDE + q4 * 4);
    async_copy16(Mb + (size_t)r * N_ + q4 * 4, Mt + r * LDS_STRIDE + q4 * 4);
  }
  async_wait0();
  __syncthreads();

  const int t  = tid & 31, w = tid >> 5;
  const int wr = w >> 1, wc = w & 1;
  const int ln = t & 15, kh = t >> 4;

  v8f acc[2][4] = {};
  for (int k0 = 0; k0 < N_; k0 += 4) {
    const int kr = k0 + 2 * kh;
    v2f af[2], bf[4];
#pragma unroll
    for (int i = 0; i < 2; ++i)
      af[i] = *(const v2f*)(Kt + (wr * 32 + i * 16 + ln) * LDS_STRIDE + kr);
#pragma unroll
    for (int j = 0; j < 4; ++j)   // B[k][m] = mem[m][k]: per-lane row, contiguous K pair
      bf[j] = *(const v2f*)(Mt + (wc * 64 + j * 16 + ln) * LDS_STRIDE + kr);
#pragma unroll
    for (int i = 0; i < 2; ++i)
#pragma unroll
      for (int j = 0; j < 4; ++j)
        acc[i][j] = wmma4(af[i], bf[j], acc[i][j]);
  }

  float* Qb = Q + (size_t)b * L_ * L_ + (size_t)l0 * L_ + m0;
#pragma unroll
  for (int i = 0; i < 2; ++i)
#pragma unroll
    for (int j = 0; j < 4; ++j)
#pragma unroll
      for (int v = 0; v < 8; ++v) {
        const int lr = wr * 32 + i * 16 + v + 8 * kh;
        const int mc = wc * 64 + j * 16 + ln;
        Qb[(size_t)lr * L_ + mc] = acc[i][j][v];
      }
}

// ---------------------------------------------------------------------------
extern "C" void kernel_launch(void* const* d_in, const int* in_sizes, int n_in,
                              void* d_out, int out_size, void* d_ws, size_t ws_size,
                              hipStream_t stream) {
  const float* key = (const float*)d_in[0];   // (32, 2048, 128)
  const float* val = (const float*)d_in[1];   // (32, 256, 128)
  const float* W   = (const float*)d_in[2];   // (128, 64, 256)
  float* out = (float*)d_out;                 // (32, 2048, 2048)

  // Workspace layout (~34.2 MiB total)
  float* W2  = (float*)d_ws;                  // 128*256
  float* C   = W2 + (size_t)N_ * A_;          // 32*128*128
  float* Mem = C + (size_t)B_ * N_ * N_;      // 32*2048*128

  k_w2<<<dim3((N_ * A_) / 256), 256, 0, stream>>>(W, W2);
  k_C<<<dim3(B_), 256, 0, stream>>>(W2, val, C);

  const size_t lds_mem = (size_t)(128 * LDS_STRIDE + 128 * 128) * sizeof(float);
  k_mem<<<dim3(L_ / 128, B_), 256, lds_mem, stream>>>(key, C, Mem);

  const size_t lds_q = (size_t)(2 * 128 * LDS_STRIDE) * sizeof(float);
  k_q<<<dim3(L_ / 128, L_ / 128, B_), 256, lds_q, stream>>>(key, Mem, out);
}